// MultiViewUniMatch_57561151701644
// MI455X (gfx1250) — compile-verified
//
#include <hip/hip_runtime.h>
#include <hip/hip_bf16.h>

typedef __attribute__((ext_vector_type(2))) float v2f;
typedef __attribute__((ext_vector_type(8))) float v8f;

#define B_W   64
#define NTOK  256
#define NH    4
#define HD    32
#define TIN   64
#define TOUT  32
#define CDIM  128
#define EPSV  1e-10f
#define SCALE 0.17677669529663687f   // 32^-0.5
#define WAVES 8

// ---------------------------------------------------------------------------
// Kernel 1: sparse progressive-focused attention. One wave32 per query.
// Phase A: lane t owns keys t and t+32 (gathered QK dots, bias, softmax,
//          PFA modulation, renorm, exact top-32 by rank counting).
// Phase B: lane d owns head-dim d (gathered AV accumulation + LEPE).
// ---------------------------------------------------------------------------
__global__ __launch_bounds__(256) void pfa_attn_kernel(
    const float* __restrict__ qkvp,        // [64,256,4,4,32]
    const float* __restrict__ pfa_values,  // [64,4,256,64]
    const int*   __restrict__ pfa_indices, // [64,4,256,64]
    const int*   __restrict__ rpi,         // [256,256]
    const float* __restrict__ bias_table,  // [961,4]
    float*       __restrict__ out_vals,    // [64,4,256,32]
    int*         __restrict__ out_idx,     // [64,4,256,32]
    float*       __restrict__ Xws)         // [64,256,128] pre-projection
{
    __shared__ __align__(16) float sh_q[WAVES][HD];
    __shared__ float sh_p[WAVES][TIN];
    __shared__ int   sh_i[WAVES][TIN];
    __shared__ float sh_kv[WAVES][TOUT];
    __shared__ int   sh_ki[WAVES][TOUT];

    const int lane = threadIdx.x & 31;
    const int wv   = threadIdx.x >> 5;
    const int qid  = blockIdx.x * WAVES + wv;       // 0 .. 65535
    const int b    = qid >> 10;                     // / (NH*NTOK)
    const int h    = (qid >> 8) & 3;
    const int i    = qid & 255;

    const float* qk_base = qkvp + (size_t)b * NTOK * 512;

    // stage scaled q row into LDS (one float per lane)
    sh_q[wv][lane] = qk_base[(size_t)i * 512 + h * HD + lane] * SCALE;
    __syncthreads();

    // ---- gathered QK dots: two keys per lane ----
    const size_t pfa_base = (((size_t)b * NH + h) * NTOK + i) * TIN;
    const int t0 = lane, t1 = lane + 32;
    const int i0 = pfa_indices[pfa_base + t0];
    const int i1 = pfa_indices[pfa_base + t1];

    const float* krow0 = qk_base + (size_t)i0 * 512 + 128 + h * HD;
    const float* krow1 = qk_base + (size_t)i1 * 512 + 128 + h * HD;
    float d0 = 0.f, d1 = 0.f;
#pragma unroll
    for (int d = 0; d < HD; d += 4) {
        float4 qv = *(const float4*)&sh_q[wv][d];           // LDS broadcast
        float4 k0 = *(const float4*)(krow0 + d);            // 128B coalesced
        float4 k1 = *(const float4*)(krow1 + d);
        d0 += qv.x * k0.x + qv.y * k0.y + qv.z * k0.z + qv.w * k0.w;
        d1 += qv.x * k1.x + qv.y * k1.y + qv.z * k1.z + qv.w * k1.w;
    }

    // relative-position bias gathered at sparse key positions
    const int e0 = rpi[i * NTOK + i0];
    const int e1 = rpi[i * NTOK + i1];
    float a0 = d0 + bias_table[e0 * NH + h];
    float a1 = d1 + bias_table[e1 * NH + h];

    // ---- softmax over 64 (2 per lane) via wave32 shuffles ----
    float m = fmaxf(a0, a1);
#pragma unroll
    for (int s = 16; s > 0; s >>= 1) m = fmaxf(m, __shfl_xor(m, s, 32));
    float ex0 = __expf(a0 - m), ex1 = __expf(a1 - m);
    float ssum = ex0 + ex1;
#pragma unroll
    for (int s = 16; s > 0; s >>= 1) ssum += __shfl_xor(ssum, s, 32);
    const float inv = 1.f / ssum;

    // progressive-focused modulation + renorm
    const float w0 = ex0 * inv * pfa_values[pfa_base + t0];
    const float w1 = ex1 * inv * pfa_values[pfa_base + t1];
    float s2 = w0 + w1;
#pragma unroll
    for (int s = 16; s > 0; s >>= 1) s2 += __shfl_xor(s2, s, 32);
    const float dn = 1.f / (s2 + EPSV);
    const float p0 = (w0 + EPSV) * dn;
    const float p1 = (w1 + EPSV) * dn;

    sh_p[wv][t0] = p0; sh_p[wv][t1] = p1;
    sh_i[wv][t0] = i0; sh_i[wv][t1] = i1;
    __syncthreads();

    // ---- exact top-32 by rank counting (matches lax.top_k stable order) ----
    int r0 = 0, r1 = 0;
#pragma unroll 8
    for (int j = 0; j < TIN; ++j) {
        const float aj = sh_p[wv][j];                        // LDS broadcast
        r0 += (aj > p0) | ((aj == p0) & (j < t0));
        r1 += (aj > p1) | ((aj == p1) & (j < t1));
    }
    const size_t ob = (((size_t)b * NH + h) * NTOK + i) * TOUT;
    if (r0 < TOUT) {
        sh_kv[wv][r0] = p0; sh_ki[wv][r0] = i0;
        out_vals[ob + r0] = p0; out_idx[ob + r0] = i0;
    }
    if (r1 < TOUT) {
        sh_kv[wv][r1] = p1; sh_ki[wv][r1] = i1;
        out_vals[ob + r1] = p1; out_idx[ob + r1] = i1;
    }
    __syncthreads();

    // ---- phase B: lane = head dim; gathered AV + LEPE ----
    float acc = 0.f;
#pragma unroll 8
    for (int r = 0; r < TOUT; ++r) {
        const float vv = sh_kv[wv][r];
        const int   ki = sh_ki[wv][r];
        acc += vv * qk_base[(size_t)ki * 512 + 256 + h * HD + lane];  // v
    }
    acc += qk_base[(size_t)i * 512 + 384 + h * HD + lane];            // v_lepe
    // X layout [b, i, h*HD+d] == row-major [16384, 128]
    Xws[((size_t)b * NTOK + i) * CDIM + h * HD + lane] = acc;
}

// ---------------------------------------------------------------------------
// Kernel 2: Y = X @ W^T + b with V_WMMA_F32_16X16X4_F32 (full f32 GEMM).
// Block = 8 waves; wave owns 16 rows x 128 cols (8 accumulator tiles).
// A layout (ISA 16x4 f32): lanes 0-15 -> K=k0,k0+1 ; lanes 16-31 -> K=k0+2,k0+3.
// W (64KB) stays L2/L1 resident across all 128 blocks.
// ---------------------------------------------------------------------------
__global__ __launch_bounds__(256) void proj_wmma_kernel(
    const float* __restrict__ X,    // [16384,128]
    const float* __restrict__ W,    // [128,128] (out_c, in_c)
    const float* __restrict__ bias, // [128]
    float*       __restrict__ Y)    // [16384,128]
{
    const int lane = threadIdx.x & 31;
    const int wv   = threadIdx.x >> 5;
    const int half = lane >> 4;           // 0: lanes 0-15, 1: lanes 16-31
    const int lm   = lane & 15;
    const int row0 = blockIdx.x * 128 + wv * 16;

    v8f acc[8] = {};
    const float* xrow = X + (size_t)(row0 + lm) * CDIM;

    for (int k0 = 0; k0 < CDIM; k0 += 4) {
        const int ka = k0 + half * 2;
        v2f a;
        a.x = xrow[ka];
        a.y = xrow[ka + 1];
#pragma unroll
        for (int nt = 0; nt < 8; ++nt) {
            const float* wrow = W + (size_t)(nt * 16 + lm) * CDIM;
            v2f bb;
            bb.x = wrow[ka];
            bb.y = wrow[ka + 1];
            acc[nt] = __builtin_amdgcn_wmma_f32_16x16x4_f32(
                /*neg_a=*/false, a, /*neg_b=*/false, bb,
                /*c_mod=*/(short)0, acc[nt],
                /*reuse_a=*/false, /*reuse_b=*/false);
        }
    }

    // epilogue: C/D layout -> VGPR j holds M = j (lanes 0-15) / M = 8+j (16-31)
#pragma unroll
    for (int nt = 0; nt < 8; ++nt) {
        const int col = nt * 16 + lm;
        const float pb = bias[col];
#pragma unroll
        for (int j = 0; j < 8; ++j) {
            const int row = row0 + half * 8 + j;
            Y[(size_t)row * CDIM + col] = acc[nt][j] + pb;
        }
    }
}

extern "C" void kernel_launch(void* const* d_in, const int* in_sizes, int n_in,
                              void* d_out, int out_size, void* d_ws, size_t ws_size,
                              hipStream_t stream) {
    const float* qkvp        = (const float*)d_in[0];
    const float* pfa_values  = (const float*)d_in[1];
    const int*   pfa_indices = (const int*)d_in[2];
    const int*   rpi         = (const int*)d_in[3];
    const float* bias_table  = (const float*)d_in[4];
    const float* proj_w      = (const float*)d_in[5];
    const float* proj_b      = (const float*)d_in[6];

    float* out      = (float*)d_out;                          // [64,256,128]
    float* out_vals = out + (size_t)B_W * NTOK * CDIM;        // [64,4,256,32]
    int*   out_idx  = (int*)(out_vals + (size_t)B_W * NH * NTOK * TOUT);
    float* Xws      = (float*)d_ws;                           // [16384,128]

    // 65536 queries / 8 waves per block
    pfa_attn_kernel<<<(B_W * NH * NTOK) / WAVES, 256, 0, stream>>>(
        qkvp, pfa_values, pfa_indices, rpi, bias_table, out_vals, out_idx, Xws);

    // 16384 rows / 128 rows per block
    proj_wmma_kernel<<<(B_W * NTOK) / 128, 256, 0, stream>>>(
        Xws, proj_w, proj_b, out);
}